// CosineDistanceLayer_24249385353609
// MI455X (gfx1250) — compile-verified
//
#include <hip/hip_runtime.h>
#include <hip/hip_bf16.h>

#define NROWS 262144
#define DCOLS 128
#define CS_BLOCKS 512          // pass-1 grid; partial buffer = 2*512*128 floats (fits in d_out)
#define PART_STRIDE 65536      // CS_BLOCKS * DCOLS

typedef __attribute__((ext_vector_type(2))) float v2f;
typedef __attribute__((ext_vector_type(8))) float v8f;

__device__ __forceinline__ void f4add(float4& a, const float4& b) {
    a.x += b.x; a.y += b.y; a.z += b.z; a.w += b.w;
}

// ---------------------------------------------------------------------------
// Pass 1: per-block column partial sums of x1 and x2 (coalesced float4 stream).
// part layout: part[mat*PART_STRIDE + block*128 + col]
// ---------------------------------------------------------------------------
__global__ __launch_bounds__(256) void colsum_kernel(
    const float* __restrict__ x1, const float* __restrict__ x2,
    float* __restrict__ part) {
    const int tid = threadIdx.x;
    const int cg  = tid & 31;   // column group: columns 4*cg .. 4*cg+3
    const int rg  = tid >> 5;   // row sub-index 0..7 (8 rows per block-iteration)
    float4 a1 = make_float4(0.f, 0.f, 0.f, 0.f);
    float4 a2 = make_float4(0.f, 0.f, 0.f, 0.f);
    const int step = CS_BLOCKS * 8;
    for (int row = blockIdx.x * 8 + rg; row < NROWS; row += step) {
        const float4 u = ((const float4*)(x1 + (size_t)row * DCOLS))[cg];
        const float4 v = ((const float4*)(x2 + (size_t)row * DCOLS))[cg];
        f4add(a1, u);
        f4add(a2, v);
    }
    __shared__ float4 l1[256];
    __shared__ float4 l2[256];
    l1[tid] = a1;
    l2[tid] = a2;
    __syncthreads();
    #pragma unroll
    for (int s = 4; s >= 1; s >>= 1) {
        if (rg < s) {
            f4add(l1[tid], l1[tid + s * 32]);
            f4add(l2[tid], l2[tid + s * 32]);
        }
        __syncthreads();
    }
    if (rg == 0) {
        float* o1 = part + (size_t)blockIdx.x * DCOLS + cg * 4;
        float* o2 = o1 + PART_STRIDE;
        *(float4*)o1 = l1[tid];
        *(float4*)o2 = l2[tid];
    }
}

// ---------------------------------------------------------------------------
// Pass 1b: reduce CS_BLOCKS partials per column -> sums[256] = {s1[128], s2[128]}
// ---------------------------------------------------------------------------
__global__ __launch_bounds__(256) void colreduce_kernel(
    const float* __restrict__ part, float* __restrict__ sums) {
    const int t   = threadIdx.x;      // 0..255
    const int mat = t >> 7;
    const int c   = t & 127;
    const float* p = part + (size_t)mat * PART_STRIDE + c;
    float a0 = 0.f, a1 = 0.f, a2 = 0.f, a3 = 0.f;
    for (int b = 0; b < CS_BLOCKS; b += 4) {
        a0 += p[(size_t)(b + 0) * DCOLS];
        a1 += p[(size_t)(b + 1) * DCOLS];
        a2 += p[(size_t)(b + 2) * DCOLS];
        a3 += p[(size_t)(b + 3) * DCOLS];
    }
    sums[t] = (a0 + a1) + (a2 + a3);
}

// ---------------------------------------------------------------------------
// Pass 2: each wave32 owns a 16-row tile.
//   num_i  : gram diagonal via V_WMMA_F32_16X16X4_F32 chain (matrix pipe, fp32)
//   v1,v2  : broadcast dots vs s1/s2 on VALU (co-executes with XDL WMMA)
// A-fragment (16x4 f32): lanes 0-15 hold cols {k0,k0+1} of rows r0..r0+15,
// lanes 16-31 hold cols {k0+2,k0+3}. Same fragment serves as B (4x16) for the
// transposed operand of the gram, so one b64 load per matrix feeds both sides.
// ---------------------------------------------------------------------------
__global__ __launch_bounds__(256) void cosine_kernel(
    const float* __restrict__ x1, const float* __restrict__ x2,
    const float* __restrict__ sums, float* __restrict__ out) {
    __shared__ float s_sum[256];  // [0..127]=s1, [128..255]=s2
    const int tid = threadIdx.x;
    s_sum[tid] = sums[tid];
    __syncthreads();

    const int wave = tid >> 5;
    const int lane = tid & 31;
    const int r0   = (blockIdx.x * 8 + wave) * 16;
    const int m    = lane & 15;
    const int kk   = (lane >> 4) * 2;          // 0 for lanes 0-15, 2 for lanes 16-31

    const float* pA = x1 + (size_t)(r0 + m) * DCOLS + kk;
    const float* pB = x2 + (size_t)(r0 + m) * DCOLS + kk;
    const float* s1 = s_sum;
    const float* s2 = s_sum + 128;

    v8f accN = {0.f, 0.f, 0.f, 0.f, 0.f, 0.f, 0.f, 0.f};
    float v1p = 0.f, v2p = 0.f;

    #pragma unroll
    for (int ks = 0; ks < 32; ++ks) {
        const int k0 = ks * 4;
        const v2f a1 = *(const v2f*)(pA + k0);
        const v2f a2 = *(const v2f*)(pB + k0);
        // gram: D = X1_tile (16x4) x X2_tile^T (4x16) + C, fp32 matrix pipe
        accN = __builtin_amdgcn_wmma_f32_16x16x4_f32(
            /*neg_a=*/false, a1, /*neg_b=*/false, a2,
            /*c_mod=*/(short)0, accN, /*reuse_a=*/false, /*reuse_b=*/false);
        // broadcast dots on VALU (co-executes with WMMA)
        v1p += a1.x * s1[k0 + kk] + a1.y * s1[k0 + kk + 1];
        v2p += a2.x * s2[k0 + kk] + a2.y * s2[k0 + kk + 1];
    }

    // Combine K halves: lane L and lane L+16 hold the two halves of row r0+(L&15).
    const float v1 = v1p + __shfl_xor(v1p, 16, 32);
    const float v2 = v2p + __shfl_xor(v2p, 16, 32);

    // Gram diagonal extraction (C/D layout: vgpr v -> M=v lanes 0-15, M=v+8 lanes 16-31):
    // row v    at (vgpr v, lane v)      (N = lane    = v)
    // row v+8  at (vgpr v, lane v+24)   (N = lane-16 = v+8)
    float numD = 0.f;
    #pragma unroll
    for (int v = 0; v < 8; ++v) {
        if (lane == v)      numD = accN[v];
        if (lane == v + 24) numD = accN[v];
    }

    int row = -1;
    if (lane < 8)       row = lane;        // rows 0..7   (v1/v2 valid: lane&15 == row)
    else if (lane >= 24) row = lane - 16;  // rows 8..15  (lane&15 == row)
    if (row >= 0) {
        out[r0 + row] = numD / sqrtf(v1 * v2);
    }
}

// ---------------------------------------------------------------------------
extern "C" void kernel_launch(void* const* d_in, const int* in_sizes, int n_in,
                              void* d_out, int out_size, void* d_ws, size_t ws_size,
                              hipStream_t stream) {
    const float* x1 = (const float*)d_in[0];
    const float* x2 = (const float*)d_in[1];
    float* out  = (float*)d_out;
    float* sums = (float*)d_ws;   // 256 floats: s1 then s2
    float* part = out;            // reuse d_out as partial scratch; fully
                                  // overwritten by cosine_kernel afterwards

    colsum_kernel<<<CS_BLOCKS, 256, 0, stream>>>(x1, x2, part);
    colreduce_kernel<<<1, 256, 0, stream>>>(part, sums);
    cosine_kernel<<<NROWS / 128, 256, 0, stream>>>(x1, x2, sums, out);
}